// ConvTemporalGraphical_38938173506051
// MI455X (gfx1250) — compile-verified
//
#include <hip/hip_runtime.h>
#include <cstddef>

// Problem constants (from the reference): N=64, V=256, C_IN=C_OUT=64, K=6 (5 used)
#define N_B   64
#define V_DIM 256
#define C_INC 64
#define C_OUT 64
#define PLANE (V_DIM * V_DIM) // 65536

typedef __attribute__((ext_vector_type(2))) float v2f;
typedef __attribute__((ext_vector_type(8))) float v8f;

static __device__ __forceinline__ v8f wmma4(v2f a, v2f b, v8f c) {
  // D(16x16,f32) = A(16x4,f32) * B(4x16,f32) + C
  return __builtin_amdgcn_wmma_f32_16x16x4_f32(false, a, false, b, (short)0, c,
                                               false, false);
}

static __device__ __forceinline__ float sigm(float t) {
  return 1.0f / (1.0f + __expf(-t));
}

static __device__ __forceinline__ float relu1(float t) {
  // single VOP2 max (constant in src0), avoids the canonicalize+max pair
  float r;
  asm("v_max_num_f32 %0, 0, %1" : "=v"(r) : "v"(t));
  return r;
}

// ---------------------------------------------------------------------------
// Kernel 1: xo_c[n,c,v] = sum_i wo[c,i]*xo[n,i,v] + bo[c]   (tiny: 0.067 G MAC)
// ---------------------------------------------------------------------------
__global__ void __launch_bounds__(256) convo_kernel(
    const float* __restrict__ xo, const float* __restrict__ wo,
    const float* __restrict__ bo, float* __restrict__ xoc) {
  int t = blockIdx.x * 256 + threadIdx.x;   // N*C_OUT*V threads
  int v = t & (V_DIM - 1);
  int c = (t >> 8) & (C_OUT - 1);
  int n = t >> 14;
  const float* xr = xo + (size_t)n * C_INC * V_DIM + v;
  const float* wr = wo + c * C_INC;
  float acc = bo[c];
#pragma unroll 8
  for (int i = 0; i < C_INC; ++i) acc = fmaf(wr[i], xr[i * V_DIM], acc);
  xoc[t] = acc;
}

// ---------------------------------------------------------------------------
// Kernel 2: fused MLP (5->16->32->64) on v_wmma_f32_16x16x4_f32.
//  - biases folded into an extra K-row (B row == 1.0), C starts at inline 0
//  - activations staged in per-wave LDS with PAIR-INTERLEAVED layout
//    ([rowpair][col][2]) so every B fragment is a single aligned ds_load_b64
//    (no cross-register shuffles), stores are ds_store_b64
//  - A_out NT-streamed with immediate offsets; fused sum_v xo_c*A_out
// ---------------------------------------------------------------------------
__global__ void __launch_bounds__(256) mlp_agg_kernel(
    const float* __restrict__ x,  const float* __restrict__ A,
    const float* __restrict__ w1, const float* __restrict__ b1,
    const float* __restrict__ w2, const float* __restrict__ b2,
    const float* __restrict__ w3, const float* __restrict__ b3,
    const float* __restrict__ xoc,
    float* __restrict__ out1, float* __restrict__ aout) {
  // per-wave LDS (v2f units): h1p[8*16] + h2p[16*16] + scolp[32] = 416 v2f
  __shared__ v2f ldsv[8 * (8 * 16 + 16 * 16 + 32)];
  const int lane = threadIdx.x & 31;
  const int wv   = threadIdx.x >> 5;
  const int half = lane >> 4;     // 0: lanes 0-15, 1: lanes 16-31
  const int m    = lane & 15;     // column / row-in-tile index
  v2f* h1v   = ldsv + wv * (8 * 16 + 16 * 16 + 32);
  v2f* h2v   = h1v + 8 * 16;
  v2f* scolv = h2v + 16 * 16;

  const int wid = blockIdx.x * 8 + wv;   // 0..1023
  const int n   = wid >> 4;              // batch
  const int w0  = (wid & 15) << 4;       // w tile start

  const v8f vzero = {0.f, 0.f, 0.f, 0.f, 0.f, 0.f, 0.f, 0.f};

  // constant bias-row B fragment: 1.0 in .x (K=base row) for lanes 0-15
  v2f bOne;
  bOne.x = half ? 0.0f : 1.0f;
  bOne.y = 0.0f;

  // ---- pre-swizzle weights into WMMA A-fragments (per-lane registers) ----
  // A 16x4 f32 layout: VGPR0 = K(0|2), VGPR1 = K(1|3); lanes 0-15 / 16-31.
  v2f a1c0, a1c1;
  a1c0.x = w1[m * 5 + (half ? 2 : 0)];
  a1c0.y = w1[m * 5 + (half ? 3 : 1)];
  {
    float t4 = w1[m * 5 + 4];            // K row 4 = real, row 5 = bias, 6,7 = 0
    float tb = b1[m];
    a1c1.x = half ? 0.0f : t4;           // K=4 | K=6(zero)
    a1c1.y = half ? 0.0f : tb;           // K=5 (bias row) | K=7(zero)
  }
  v2f a2[2][4], a2b[2];
#pragma unroll
  for (int mt = 0; mt < 2; ++mt) {
#pragma unroll
    for (int k = 0; k < 4; ++k) {
      a2[mt][k].x = w2[(mt * 16 + m) * 16 + 4 * k + (half ? 2 : 0)];
      a2[mt][k].y = w2[(mt * 16 + m) * 16 + 4 * k + (half ? 3 : 1)];
    }
    float tb = b2[mt * 16 + m];
    a2b[mt].x = half ? 0.0f : tb;        // bias K-chunk (row 16)
    a2b[mt].y = 0.0f;
  }
  v2f a3[4][8], a3b[4];
#pragma unroll
  for (int mt = 0; mt < 4; ++mt) {
#pragma unroll
    for (int k = 0; k < 8; ++k) {
      a3[mt][k].x = w3[(mt * 16 + m) * 32 + 4 * k + (half ? 2 : 0)];
      a3[mt][k].y = w3[(mt * 16 + m) * 32 + 4 * k + (half ? 3 : 1)];
    }
    float tb = b3[mt * 16 + m];
    a3b[mt].x = half ? 0.0f : tb;        // bias K-chunk (row 32)
    a3b[mt].y = 0.0f;
  }

  float acc[4][8];
#pragma unroll
  for (int mt = 0; mt < 4; ++mt)
#pragma unroll
    for (int r = 0; r < 8; ++r) acc[mt][r] = 0.0f;

  // ---- loop-carried pointers (advanced by one v-row per iteration) ----
  const float* ap = A + (size_t)n * 6 * PLANE + w0 + m;        // + i*PLANE
  const float* xp = xoc + (size_t)n * C_OUT * V_DIM;           // + c*V + v
  // 4 per-mt output bases; stores use r*PLANE immediate offsets (< 2^23).
  float* op[4];
#pragma unroll
  for (int mt = 0; mt < 4; ++mt)
    op[mt] = aout + (size_t)(n * C_OUT + mt * 16 + half * 8) * PLANE + w0 + m;

#pragma unroll 1
  for (int v = 0; v < V_DIM; ++v) {
    // stage xo_c[:, v] column: lane loads channel pair (2*lane, 2*lane+1)
    {
      v2f sc;
      sc.x = xp[(2 * lane) * V_DIM];
      sc.y = xp[(2 * lane + 1) * V_DIM];
      scolv[lane] = sc;                  // one ds_store_b64
    }

    // layer-1 B fragments straight from A (rows = input channels 0..4)
#pragma unroll
    for (int i = 0; i < 5; ++i)
      __builtin_prefetch(ap + (size_t)i * PLANE + V_DIM, 0, 3);  // next v row
    v2f bb0, bb1;
    bb0.x = __builtin_nontemporal_load(ap + (size_t)(half ? 2 : 0) * PLANE);
    bb0.y = __builtin_nontemporal_load(ap + (size_t)(half ? 3 : 1) * PLANE);
    {
      float t4 = __builtin_nontemporal_load(ap + (size_t)4 * PLANE);
      bb1.x = half ? 0.0f : t4;          // K=4 input | K=6 zero
      bb1.y = half ? 0.0f : 1.0f;        // K=5 bias row | K=7 zero
    }

    // ---- layer 1: 5(+bias) -> 16, C starts at inline 0 ----
    v8f d1 = wmma4(a1c0, bb0, vzero);
    d1 = wmma4(a1c1, bb1, d1);
    // pair-interleaved stash: rowpair = 4*half + j, cols = m
#pragma unroll
    for (int j = 0; j < 4; ++j) {
      v2f t;
      t.x = relu1(d1[2 * j]);
      t.y = relu1(d1[2 * j + 1]);
      h1v[(4 * half + j) * 16 + m] = t;  // ds_store_b64
    }

    // ---- layer 2: 16(+bias) -> 32 ----
    v8f d2[2];
    d2[0] = wmma4(a2b[0], bOne, vzero);  // bias chunk
    d2[1] = wmma4(a2b[1], bOne, vzero);
#pragma unroll
    for (int k = 0; k < 4; ++k) {
      v2f bb = h1v[(2 * k + half) * 16 + m];   // rows 4k+2h, 4k+2h+1
      d2[0] = wmma4(a2[0][k], bb, d2[0]);
      d2[1] = wmma4(a2[1][k], bb, d2[1]);
    }
#pragma unroll
    for (int mt = 0; mt < 2; ++mt)
#pragma unroll
      for (int j = 0; j < 4; ++j) {
        v2f t;
        t.x = relu1(d2[mt][2 * j]);
        t.y = relu1(d2[mt][2 * j + 1]);
        h2v[(mt * 8 + 4 * half + j) * 16 + m] = t;
      }

    // ---- layer 3: 32(+bias) -> 64 ----
    v8f d3[4];
#pragma unroll
    for (int mt = 0; mt < 4; ++mt) d3[mt] = wmma4(a3b[mt], bOne, vzero);
#pragma unroll
    for (int k = 0; k < 8; ++k) {
      v2f bb = h2v[(2 * k + half) * 16 + m];   // rows 4k+2h, 4k+2h+1
#pragma unroll
      for (int mt = 0; mt < 4; ++mt) d3[mt] = wmma4(a3[mt][k], bb, d3[mt]);
    }

    // sigmoid -> streaming A_out store (NT, imm offsets) + fused reduction
#pragma unroll
    for (int mt = 0; mt < 4; ++mt)
#pragma unroll
      for (int j = 0; j < 4; ++j) {
        float a0 = sigm(d3[mt][2 * j]);
        float a1 = sigm(d3[mt][2 * j + 1]);
        __builtin_nontemporal_store(a0, op[mt] + (size_t)(2 * j) * PLANE);
        __builtin_nontemporal_store(a1, op[mt] + (size_t)(2 * j + 1) * PLANE);
        v2f sc = scolv[mt * 8 + 4 * half + j];   // channels c, c+1
        acc[mt][2 * j]     = fmaf(sc.x, a0, acc[mt][2 * j]);
        acc[mt][2 * j + 1] = fmaf(sc.y, a1, acc[mt][2 * j + 1]);
      }

    // advance loop-carried pointers by one v-row
    ap += V_DIM;
    xp += 1;
#pragma unroll
    for (int mt = 0; mt < 4; ++mt) op[mt] += V_DIM;
  }

  // epilogue: posterior_prob = sigmoid(acc); out1 = x * pp
#pragma unroll
  for (int mt = 0; mt < 4; ++mt)
#pragma unroll
    for (int r = 0; r < 8; ++r) {
      int c = mt * 16 + r + half * 8;
      size_t idx = ((size_t)n * C_OUT + c) * V_DIM + w0 + m;
      out1[idx] = x[idx] * sigm(acc[mt][r]);
    }
}

extern "C" void kernel_launch(void* const* d_in, const int* in_sizes, int n_in,
                              void* d_out, int out_size, void* d_ws,
                              size_t ws_size, hipStream_t stream) {
  (void)in_sizes; (void)n_in; (void)out_size; (void)ws_size;
  const float* x  = (const float*)d_in[0];
  const float* xo = (const float*)d_in[1];
  const float* A  = (const float*)d_in[2];
  const float* w1 = (const float*)d_in[3];
  const float* b1 = (const float*)d_in[4];
  const float* w2 = (const float*)d_in[5];
  const float* b2 = (const float*)d_in[6];
  const float* w3 = (const float*)d_in[7];
  const float* b3 = (const float*)d_in[8];
  const float* wo = (const float*)d_in[9];
  const float* bo = (const float*)d_in[10];

  float* out1 = (float*)d_out;                               // [N,C_OUT,V]
  float* aout = out1 + (size_t)N_B * C_OUT * V_DIM;          // [N,64,V,V]
  float* xoc  = (float*)d_ws;                                // 4 MB scratch

  convo_kernel<<<(N_B * C_OUT * V_DIM) / 256, 256, 0, stream>>>(xo, wo, bo, xoc);
  mlp_agg_kernel<<<128, 256, 0, stream>>>(x, A, w1, b1, w2, b2, w3, b3, xoc,
                                          out1, aout);
}